// IIR_43937515438318
// MI455X (gfx1250) — compile-verified
//
#include <hip/hip_runtime.h>
#include <hip/hip_bf16.h>
#include <stdint.h>

// B=256, T=8192, F=64, C=16, K=32
#define BB 256
#define TT 8192

typedef float v2f __attribute__((ext_vector_type(2)));
typedef float v8f __attribute__((ext_vector_type(8)));

// ---------------------------------------------------------------------------
// f32 WMMA 16x16 tile GEMM: D += A(16xK, LDS) * W(KxN, global), K stepped by 4.
// Fragment layouts per CDNA5 ISA (wave32):
//   A 16x4 : lane(m=l%16), vgpr r -> k = 2*(l>=16)+r
//   B 4x16 : lane(n=l%16), vgpr r -> k = 2*(l>=16)+r
//   C 16x16: lane(n=l%16), vgpr r -> m = r + 8*(l>=16)
// ---------------------------------------------------------------------------
__device__ __forceinline__ v8f gemm_tile_f32(const float* __restrict__ sA, int lda,
                                             const float* __restrict__ gW, int ldb,
                                             int n0, int Ktot, int l16, int lhalf,
                                             int nmax) {
  v8f acc = {};
  const int n = n0 + l16;
  for (int k0 = 0; k0 < Ktot; k0 += 4) {
    const int ka = k0 + 2 * lhalf;
    v2f a, b;
    a.x = sA[l16 * lda + ka];
    a.y = sA[l16 * lda + ka + 1];
    b.x = (n < nmax) ? gW[ka * ldb + n] : 0.0f;
    b.y = (n < nmax) ? gW[(ka + 1) * ldb + n] : 0.0f;
    acc = __builtin_amdgcn_wmma_f32_16x16x4_f32(false, a, false, b,
                                                (short)0, acc, false, false);
  }
  return acc;
}

// ---------------------------------------------------------------------------
// Kernel 1: per-batch coefficient MLP via fp32 WMMA.
// One wave (32 threads) per 16-row batch tile; 16 workgroups total.
// Produces coef[b][0..31] = b_coeffs (b0, filt[0..30]),
//          coef[b][32..63] = a_coeffs (filt[31..62]).
// ---------------------------------------------------------------------------
__global__ __launch_bounds__(32) void coef_kernel(
    const float* __restrict__ c,        // B x 16
    const float* __restrict__ pre0_w,   // 16 x 64
    const float* __restrict__ pre0_b,   // 64
    const float* __restrict__ alpha,    // 64
    const float* __restrict__ film_w,   // 16 x 128
    const float* __restrict__ film_b,   // 128
    const float* __restrict__ glu_w,    // 64 x 128
    const float* __restrict__ glu_b,    // 128
    const float* __restrict__ filt_w,   // 64 x 63
    const float* __restrict__ filt_b,   // 63
    const float* __restrict__ b0,       // B
    float* __restrict__ coef)           // B x 64
{
  __shared__ float sA[16 * 16];    // input c tile
  __shared__ float sH[16 * 64];    // h1 / h2 / final filt
  __shared__ float sG[16 * 128];   // gb / ag
  __shared__ float sF[16 * 64];    // h3 (GLU output)

  const int lane  = threadIdx.x;
  const int l16   = lane & 15;
  const int lhalf = lane >> 4;
  const int m0    = blockIdx.x * 16;

  for (int idx = lane; idx < 256; idx += 32)
    sA[idx] = c[(m0 + (idx >> 4)) * 16 + (idx & 15)];
  __syncthreads();

  // ---- layer 1: h1 = PReLU(c @ pre0_w + pre0_b) ----
  for (int nt = 0; nt < 4; ++nt) {
    v8f acc = gemm_tile_f32(sA, 16, pre0_w, 64, nt * 16, 16, l16, lhalf, 64);
    const int n = nt * 16 + l16;
    const float bia = pre0_b[n], al = alpha[n];
#pragma unroll
    for (int r = 0; r < 8; ++r) {
      float v = acc[r] + bia;
      sH[(r + 8 * lhalf) * 64 + n] = (v > 0.0f) ? v : al * v;
    }
  }
  // ---- FiLM params: gb = c @ film_w + film_b ----
  for (int nt = 0; nt < 8; ++nt) {
    v8f acc = gemm_tile_f32(sA, 16, film_w, 128, nt * 16, 16, l16, lhalf, 128);
    const int n = nt * 16 + l16;
    const float bia = film_b[n];
#pragma unroll
    for (int r = 0; r < 8; ++r)
      sG[(r + 8 * lhalf) * 128 + n] = acc[r] + bia;
  }
  __syncthreads();
  // ---- FiLM: h2 = gamma*h1 + beta ----
  for (int idx = lane; idx < 16 * 64; idx += 32) {
    const int m = idx >> 6, n = idx & 63;
    sH[idx] = sG[m * 128 + n] * sH[idx] + sG[m * 128 + 64 + n];
  }
  __syncthreads();
  // ---- GLU matmul: ag = h2 @ glu_w + glu_b ----
  for (int nt = 0; nt < 8; ++nt) {
    v8f acc = gemm_tile_f32(sH, 64, glu_w, 128, nt * 16, 64, l16, lhalf, 128);
    const int n = nt * 16 + l16;
    const float bia = glu_b[n];
#pragma unroll
    for (int r = 0; r < 8; ++r)
      sG[(r + 8 * lhalf) * 128 + n] = acc[r] + bia;
  }
  __syncthreads();
  // ---- GLU activation: h3 = a * sigmoid(g) ----
  for (int idx = lane; idx < 16 * 64; idx += 32) {
    const int m = idx >> 6, n = idx & 63;
    const float a = sG[m * 128 + n], g = sG[m * 128 + 64 + n];
    sF[idx] = a * (1.0f / (1.0f + __expf(-g)));
  }
  __syncthreads();
  // ---- filters: filt = softsign(h3 @ filt_w + filt_b), 63 cols (padded to 64)
  for (int nt = 0; nt < 4; ++nt) {
    v8f acc = gemm_tile_f32(sF, 64, filt_w, 63, nt * 16, 64, l16, lhalf, 63);
    const int n = nt * 16 + l16;
    const float bia = (n < 63) ? filt_b[n] : 0.0f;
#pragma unroll
    for (int r = 0; r < 8; ++r) {
      float v = acc[r] + bia;
      sH[(r + 8 * lhalf) * 64 + n] = v / (1.0f + fabsf(v));
    }
  }
  __syncthreads();
  // ---- pack coefficients: coef[b][0]=b0, coef[b][k]=filt[k-1] for k=1..63 ----
  for (int idx = lane; idx < 16 * 64; idx += 32) {
    const int m = idx >> 6, k = idx & 63;
    const int bg = m0 + m;
    coef[bg * 64 + k] = (k == 0) ? b0[bg] : sH[m * 64 + (k - 1)];
  }
}

// ---------------------------------------------------------------------------
// Async global -> LDS single-dword load (CDNA5, per-lane, tracked by ASYNCcnt)
// ---------------------------------------------------------------------------
__device__ __forceinline__ void async_load_f32(const float* gaddr, float* lptr) {
  unsigned loff = (unsigned)(uintptr_t)lptr;  // low 32 bits = LDS byte address
  asm volatile("global_load_async_to_lds_b32 %0, %1, off"
               :: "v"(loff), "v"(gaddr) : "memory");
}
__device__ __forceinline__ void wait_asynccnt0() {
  asm volatile("s_wait_asynccnt 0" ::: "memory");
}

// ---------------------------------------------------------------------------
// Kernel 2: blocked IIR scan, one wave32 per batch.
//   y_t = sum_{k=0..31} b_k x_{t-k} - sum_{k=1..32} a_k y_{t-k}
// Block of L=64:  u_i = f_i - sum_{k>i} a_k s_{k-1-i}  (i<32), u_i = f_i else;
//                 y_i = sum_{j<=i} h_{i-j} u_j   (h = AR impulse response)
// ---------------------------------------------------------------------------
__global__ __launch_bounds__(32) void iir_kernel(
    const float* __restrict__ x,     // B x T
    const float* __restrict__ coef,  // B x 64
    float* __restrict__ y)           // B x T
{
  __shared__ float hpad[128];  // hpad[64+n] = h_n (n=0..63), zeros below
  __shared__ float xb[96];     // sliding x window: xb[32+i] = x[t0+i]
  __shared__ float ub[64];
  __shared__ float yb[64];
  __shared__ float sbuf[32];   // state s_j = y[t0-1-j]
  __shared__ float ca[32];     // a_{k+1} = coef[32+k]
  __shared__ float cb[32];     // b_k     = coef[k]

  const int lane = threadIdx.x;
  const int b    = blockIdx.x;

  cb[lane] = coef[b * 64 + lane];
  ca[lane] = coef[b * 64 + 32 + lane];
  hpad[lane] = 0.0f; hpad[32 + lane] = 0.0f;
  hpad[64 + lane] = 0.0f; hpad[96 + lane] = 0.0f;
  sbuf[lane] = 0.0f;
  xb[lane] = 0.0f;  // x_{-32..-1} = 0
  __syncthreads();

  const float av = ca[lane];  // this lane's a_{lane+1}
  if (lane == 0) hpad[64] = 1.0f;  // h_0
  __syncthreads();
  // h_n = -sum_{k=1..min(n,32)} a_k h_{n-k}; zero padding handles k>n
  for (int n = 1; n < 64; ++n) {
    float term = -av * hpad[64 + n - 1 - lane];
    for (int off = 16; off; off >>= 1)
      term += __shfl_xor(term, off, 32);
    if (lane == 0) hpad[64 + n] = term;
    __syncthreads();
  }

  const float* xg = x + (size_t)b * TT;
  float* yg = y + (size_t)b * TT;

  for (int blk = 0; blk < 128; ++blk) {
    const int t0 = blk * 64;
    // stream next 64 samples into LDS via async DMA path
    async_load_f32(xg + t0 + lane,      &xb[32 + lane]);
    async_load_f32(xg + t0 + 32 + lane, &xb[64 + lane]);
    if (t0 + 512 < TT) __builtin_prefetch(xg + t0 + 512, 0, 1);
    wait_asynccnt0();
    __syncthreads();

    // FIR: f_i = sum_k b_k x[t0+i-k], lane handles i=lane and i=lane+32
    float f0 = 0.0f, f1 = 0.0f;
#pragma unroll
    for (int k = 0; k < 32; ++k) {
      const float bk = cb[k];
      f0 = fmaf(bk, xb[32 + lane - k], f0);
      f1 = fmaf(bk, xb[64 + lane - k], f1);
    }
    // state correction (only i<32): u_i = f_i - sum_{j=i..31} a_{j+1} s_{j-i}
    float corr = 0.0f;
    for (int j = lane; j < 32; ++j)
      corr = fmaf(ca[j], sbuf[j - lane], corr);
    ub[lane]      = f0 - corr;
    ub[32 + lane] = f1;
    __syncthreads();

    // Toeplitz matvec: y_i = sum_j h_{i-j} u_j (zero-padded h)
    float acc0 = 0.0f, acc1 = 0.0f;
#pragma unroll
    for (int j = 0; j < 64; ++j) {
      const float u = ub[j];
      acc0 = fmaf(hpad[64 + lane - j], u, acc0);  // i = lane
      acc1 = fmaf(hpad[96 + lane - j], u, acc1);  // i = lane+32
    }
    yb[lane] = acc0; yb[32 + lane] = acc1;
    yg[t0 + lane] = acc0;
    yg[t0 + 32 + lane] = acc1;
    __syncthreads();

    // next state s_j = y[t0+63-j]; slide x window
    const float snew  = yb[63 - lane];
    const float xtail = xb[64 + lane];
    __syncthreads();
    sbuf[lane] = snew;
    xb[lane] = xtail;
    __syncthreads();
  }
}

extern "C" void kernel_launch(void* const* d_in, const int* in_sizes, int n_in,
                              void* d_out, int out_size, void* d_ws, size_t ws_size,
                              hipStream_t stream) {
  const float* x        = (const float*)d_in[0];   // (B,T,1)
  const float* c        = (const float*)d_in[1];   // (B,1,16)
  const float* pre0_w   = (const float*)d_in[2];   // (16,64)
  const float* pre0_b   = (const float*)d_in[3];   // (64,)
  const float* alpha    = (const float*)d_in[4];   // (64,)
  const float* film_w   = (const float*)d_in[5];   // (16,128)
  const float* film_b   = (const float*)d_in[6];   // (128,)
  const float* glu_w    = (const float*)d_in[7];   // (64,128)
  const float* glu_b    = (const float*)d_in[8];   // (128,)
  const float* filt_w   = (const float*)d_in[9];   // (64,63)
  const float* filt_b   = (const float*)d_in[10];  // (63,)
  const float* b0       = (const float*)d_in[11];  // (B,)

  float* coef = (float*)d_ws;        // B x 64 floats
  float* out  = (float*)d_out;       // B x T

  coef_kernel<<<BB / 16, 32, 0, stream>>>(c, pre0_w, pre0_b, alpha, film_w,
                                          film_b, glu_w, glu_b, filt_w, filt_b,
                                          b0, coef);
  iir_kernel<<<BB, 32, 0, stream>>>(x, coef, out);
}